// YOLOLayer2_56556129354016
// MI455X (gfx1250) — compile-verified
//
#include <hip/hip_runtime.h>

#define GDIM 52
#define NA 3
#define NB 32
#define NCLS 80
#define GG (GDIM * GDIM)
#define NCELL (NB * NA * GG)   // 259584
#define CH (NCLS + 5)          // 85
#define BLK 256
#define NQ 9                   // x,y,w,h, conf_obj, conf_noobj, cls, n_obj, n_noobj
#define NBLK ((NCELL + BLK - 1) / BLK)  // 1014
#define EPSV 1e-12f

typedef __attribute__((ext_vector_type(2))) float v2f;
typedef __attribute__((ext_vector_type(8))) float v8f;

__device__ __forceinline__ float sigmoidf_fast(float z) {
    return 1.0f / (1.0f + __expf(-z));
}

// torch BCELoss on already-sigmoided prob, with clip to [EPS, 1]
__device__ __forceinline__ float bce_clip(float p, float t) {
    float pc = fminf(fmaxf(p, EPSV), 1.0f);
    float qc = fminf(fmaxf(1.0f - p, EPSV), 1.0f);
    return -(t * __logf(pc) + (1.0f - t) * __logf(qc));
}

// Wave32 sum via V_WMMA_F32_16X16X4_F32 with B = ones(4x16).
// A layout (ISA 7.12.2, 32-bit A 16x4): lane L contributes A[L%16][L<16?0:2] (vgpr0)
// and A[L%16][L<16?1:3] (vgpr1). With a1=0: D[m][n] = v(m) + v(m+16).
// D layout: vgpr r holds M=r (lanes 0-15) / M=8+r (lanes 16-31). Summing the 8
// D vgprs per lane gives half-sums; one shfl_xor(16) completes the 32-lane sum.
__device__ __forceinline__ float wave_sum_wmma(float v) {
    v2f a; a[0] = v;    a[1] = 0.0f;
    v2f b; b[0] = 1.0f; b[1] = 1.0f;
    v8f c = {};
    c = __builtin_amdgcn_wmma_f32_16x16x4_f32(false, a, false, b,
                                              (short)0, c, false, false);
    float s = c[0] + c[1] + c[2] + c[3] + c[4] + c[5] + c[6] + c[7];
    s += __shfl_xor(s, 16, 32);
    return s;
}

__global__ __launch_bounds__(BLK)
void yolo_main(const float* __restrict__ x,
               const float* __restrict__ tx, const float* __restrict__ ty,
               const float* __restrict__ tw, const float* __restrict__ th,
               const float* __restrict__ tconf, const float* __restrict__ tcls,
               const float* __restrict__ sw1,
               const int* __restrict__ obj, const int* __restrict__ noobj,
               const int* __restrict__ img_dim,
               float* __restrict__ out, float* __restrict__ bpart) {
    const int cell = blockIdx.x * BLK + threadIdx.x;

    float p[NQ];
#pragma unroll
    for (int q = 0; q < NQ; ++q) p[q] = 0.0f;

    if (cell < NCELL) {
        const int j = cell % GDIM;
        const int i = (cell / GDIM) % GDIM;
        const int a = (cell / GG) % NA;
        const int b = cell / (NA * GG);

        const float stride = (float)img_dim[0] / (float)GDIM;  // 8.0
        const float AW[NA] = {10.0f, 16.0f, 33.0f};
        const float AH[NA] = {13.0f, 30.0f, 23.0f};

        const int xbase = (b * (NA * CH) + a * CH) * GG + i * GDIM + j;
        const int obase = cell * CH;

        const float zx = __builtin_nontemporal_load(x + xbase + 0 * GG);
        const float zy = __builtin_nontemporal_load(x + xbase + 1 * GG);
        const float zw = __builtin_nontemporal_load(x + xbase + 2 * GG);
        const float zh = __builtin_nontemporal_load(x + xbase + 3 * GG);
        const float zc = __builtin_nontemporal_load(x + xbase + 4 * GG);

        const float px = sigmoidf_fast(zx);
        const float py = sigmoidf_fast(zy);
        const float pw = zw;
        const float ph = zh;
        const float pconf = sigmoidf_fast(zc);

        // output: pred_boxes * stride, conf, cls
        __builtin_nontemporal_store((px + (float)j) * stride, out + obase + 0);
        __builtin_nontemporal_store((py + (float)i) * stride, out + obase + 1);
        __builtin_nontemporal_store(__expf(pw) * (AW[a] / stride) * stride, out + obase + 2);
        __builtin_nontemporal_store(__expf(ph) * (AH[a] / stride) * stride, out + obase + 3);
        __builtin_nontemporal_store(pconf, out + obase + 4);

        const float om = obj[cell]   ? 1.0f : 0.0f;
        const float nm = noobj[cell] ? 1.0f : 0.0f;
        const float w1 = sw1[cell];

        const float dx = px - tx[cell];
        const float dy = py - ty[cell];
        const float dw = pw - tw[cell];
        const float dh = ph - th[cell];
        p[0] = om * dx * dx * w1;
        p[1] = om * dy * dy * w1;
        p[2] = om * dw * dw * w1;
        p[3] = om * dh * dh * w1;

        const float bc = bce_clip(pconf, tconf[cell]);
        p[4] = om * bc * w1;
        p[5] = nm * bc;
        p[7] = om;
        p[8] = nm;

        const float* tc = tcls + (size_t)cell * NCLS;
        float cls = 0.0f;
        for (int c = 0; c < NCLS; ++c) {
            const float z = __builtin_nontemporal_load(x + xbase + (5 + c) * GG);
            const float pc = sigmoidf_fast(z);
            __builtin_nontemporal_store(pc, out + obase + 5 + c);
            const float t = __builtin_nontemporal_load(tc + c);
            cls += bce_clip(pc, t);
        }
        p[6] = om * cls;
    }

    // ---- block reduction: WMMA per wave, then LDS across the 8 waves ----
    __shared__ float wsum[NQ][BLK / 32];
    const int lane = threadIdx.x & 31;
    const int wid = threadIdx.x >> 5;
#pragma unroll
    for (int q = 0; q < NQ; ++q) {
        const float s = wave_sum_wmma(p[q]);   // EXEC all-1s here
        if (lane == 0) wsum[q][wid] = s;
    }
    __syncthreads();
    if (threadIdx.x < NQ) {
        float s = 0.0f;
#pragma unroll
        for (int w = 0; w < BLK / 32; ++w) s += wsum[threadIdx.x][w];
        bpart[blockIdx.x * NQ + threadIdx.x] = s;
    }
}

__global__ __launch_bounds__(BLK)
void yolo_finalize(const float* __restrict__ bpart, float* __restrict__ out) {
    __shared__ float sd[NQ][BLK];
    const int t = threadIdx.x;
    float acc[NQ];
#pragma unroll
    for (int q = 0; q < NQ; ++q) acc[q] = 0.0f;
    for (int blk = t; blk < NBLK; blk += BLK)
        for (int q = 0; q < NQ; ++q) acc[q] += bpart[blk * NQ + q];
#pragma unroll
    for (int q = 0; q < NQ; ++q) sd[q][t] = acc[q];
    __syncthreads();
    for (int off = BLK / 2; off > 0; off >>= 1) {
        if (t < off)
            for (int q = 0; q < NQ; ++q) sd[q][t] += sd[q][t + off];
        __syncthreads();
    }
    if (t == 0) {
        const float n_obj = fmaxf(sd[7][0], 1.0f);
        const float n_noobj = fmaxf(sd[8][0], 1.0f);
        const float loss =
            (sd[0][0] + sd[1][0] + sd[2][0] + sd[3][0] + sd[4][0]) / n_obj +
            100.0f * sd[5][0] / n_noobj +
            sd[6][0] / (n_obj * (float)NCLS);
        out[(size_t)NCELL * CH] = loss;
    }
}

extern "C" void kernel_launch(void* const* d_in, const int* in_sizes, int n_in,
                              void* d_out, int out_size, void* d_ws, size_t ws_size,
                              hipStream_t stream) {
    const float* x     = (const float*)d_in[0];
    const float* tx    = (const float*)d_in[1];
    const float* ty    = (const float*)d_in[2];
    const float* tw    = (const float*)d_in[3];
    const float* th    = (const float*)d_in[4];
    const float* tconf = (const float*)d_in[5];
    const float* tcls  = (const float*)d_in[6];
    const float* sw1   = (const float*)d_in[7];
    const int* obj     = (const int*)d_in[8];
    const int* noobj   = (const int*)d_in[9];
    const int* img_dim = (const int*)d_in[10];

    float* out   = (float*)d_out;
    float* bpart = (float*)d_ws;   // NBLK * NQ floats (~36.5 KB)

    yolo_main<<<dim3(NBLK), dim3(BLK), 0, stream>>>(
        x, tx, ty, tw, th, tconf, tcls, sw1, obj, noobj, img_dim, out, bpart);
    yolo_finalize<<<dim3(1), dim3(BLK), 0, stream>>>(bpart, out);
}